// FeaturePropagation_84945863181017
// MI455X (gfx1250) — compile-verified
//
#include <hip/hip_runtime.h>

// Problem constants (from reference setup_inputs)
#define BLPAIRS 16      // B*L = 2*8
#define NPTS    8192    // original points
#define NSUB    2048    // subsampled points
#define DIM     256     // feature dim
#define KNN     3
#define EPS_W   1e-6f
#define TILE_M  128     // original points per workgroup (8 waves * 16 rows)
#define BLOCK   256
#define WAVES   8
#define NTILES  (NSUB / 16)   // 128 column tiles of 16

typedef __attribute__((ext_vector_type(2))) float v2f;
typedef __attribute__((ext_vector_type(8))) float v8f;
typedef unsigned long long u64;
typedef unsigned int u32;

static __device__ __forceinline__ u64 umin64(u64 a, u64 b) { return a < b ? a : b; }
static __device__ __forceinline__ u64 umax64(u64 a, u64 b) { return a > b ? a : b; }

__global__ __launch_bounds__(BLOCK)
void fp_knn_wmma_kernel(const float* __restrict__ xyz_sub,
                        const float* __restrict__ feats_sub,
                        const float* __restrict__ xyz_org,
                        float* __restrict__ out)
{
    // Overlay region: phase 1 uses it as packed sub points (32KB),
    // after a barrier it becomes the per-row top-3 merge scratch (48KB).
    __shared__ __align__(16) unsigned char smem[48 * 1024];
    __shared__ float s_w[TILE_M * KNN];
    __shared__ int   s_i[TILE_M * KNN];

    float4* s_sub = (float4*)smem;          // [NSUB] (-2x,-2y,-2z,|b|^2)
    u64*    s_mk  = (u64*)smem;             // [TILE_M*48] packed keys (48KB)

    const int bl     = blockIdx.y;
    const int tid    = threadIdx.x;
    const int wave   = tid >> 5;
    const int lane   = tid & 31;
    const int laneLo = lane & 15;
    const bool hiHalf = (lane >= 16);

    const float* sub_xyz = xyz_sub   + (size_t)bl * NSUB * 3;
    const float* org_xyz = xyz_org   + (size_t)bl * NPTS * 3;
    const float* feats   = feats_sub + (size_t)bl * NSUB * DIM;
    float*       outp    = out       + (size_t)bl * NPTS * DIM;

    // ---- Phase 0: stage subsampled points into LDS, pre-scaled by -2,
    //      with |b|^2 in .w
    for (int p = tid; p < NSUB; p += BLOCK) {
        float bx = sub_xyz[p * 3 + 0];
        float by = sub_xyz[p * 3 + 1];
        float bz = sub_xyz[p * 3 + 2];
        s_sub[p] = make_float4(-2.0f * bx, -2.0f * by, -2.0f * bz,
                               bx * bx + by * by + bz * bz);
    }
    __syncthreads();

    // ---- Phase 1: squared distances via V_WMMA_F32_16X16X4_F32.
    // A[m,:] = [ax, ay, az, |a|^2], B[:,n] = [-2bx, -2by, -2bz, 1],
    // C[m,n] = |b_n|^2  ==>  D[m,n] = |a|^2 + |b|^2 - 2 a.b
    const int m = blockIdx.x * TILE_M + wave * 16 + laneLo;
    const float ax = org_xyz[m * 3 + 0];
    const float ay = org_xyz[m * 3 + 1];
    const float az = org_xyz[m * 3 + 2];
    const float a2 = ax * ax + ay * ay + az * az;

    // 16x4 f32 A layout: lanes 0-15 hold (K0,K1), lanes 16-31 hold (K2,K3)
    v2f Af;
    Af.x = hiHalf ? az : ax;
    Af.y = hiHalf ? a2 : ay;

    // per-lane top-3 keys per row slot: key = (bits(max(d2,0)) << 32) | n
    u64 K0[8], K1[8], K2[8];
#pragma unroll
    for (int v = 0; v < 8; ++v) K0[v] = K1[v] = K2[v] = 0xFFFFFFFFFFFFFFFFull;

    // software pipeline the LDS point fetch
    float4 p = s_sub[laneLo];
    for (int t = 0; t < NTILES; ++t) {
        const u64 ncur = (u64)(t * 16 + laneLo);
        const int tn = (t + 1 < NTILES) ? (t + 1) : t;
        float4 pn = s_sub[tn * 16 + laneLo];

        // 4x16 f32 B layout mirrors A: lanes 0-15 (K0,K1), lanes 16-31 (K2,K3)
        v2f Bf;
        Bf.x = hiHalf ? p.z : p.x;
        Bf.y = hiHalf ? 1.0f : p.y;

        v8f Cf;
#pragma unroll
        for (int v = 0; v < 8; ++v) Cf[v] = p.w;   // column-broadcast |b_n|^2

        v8f D = __builtin_amdgcn_wmma_f32_16x16x4_f32(
            false, Af, false, Bf, (short)0, Cf, false, false);

#pragma unroll
        for (int v = 0; v < 8; ++v) {
            // Integer-domain clamp: for non-NaN d, smax(bits(d),0) is
            // bit-identical to bits(fmax(d,0)) -> single v_max_i32
            int hb = __float_as_int(D[v]);
            hb = hb < 0 ? 0 : hb;
            u64 k = ((u64)(u32)hb << 32) | ncur;
            // branchless sorted insert (keep 3 smallest keys): v_min/max_u64
            K2[v] = umin64(K2[v], umax64(K1[v], k));
            K1[v] = umin64(K1[v], umax64(K0[v], k));
            K0[v] = umin64(K0[v], k);
        }
        p = pn;
    }

    // ---- Phase 2: dump per-lane top-3 keys to LDS (reuses sub-point region)
    __syncthreads();   // everyone done reading s_sub
#pragma unroll
    for (int v = 0; v < 8; ++v) {
        const int row  = wave * 16 + (hiHalf ? v + 8 : v);
        const int base = row * 48 + laneLo * 3;
        s_mk[base + 0] = K0[v];
        s_mk[base + 1] = K1[v];
        s_mk[base + 2] = K2[v];
    }
    __syncthreads();

    // ---- Phase 3: exact global top-3 per row + normalized weights
    if (tid < TILE_M) {
        u64 G0 = 0xFFFFFFFFFFFFFFFFull, G1 = G0, G2 = G0;
        const int base = tid * 48;
        for (int j = 0; j < 48; ++j) {
            u64 k = s_mk[base + j];
            G2 = umin64(G2, umax64(G1, k));
            G1 = umin64(G1, umax64(G0, k));
            G0 = umin64(G0, k);
        }
        const float d0 = __uint_as_float((u32)(G0 >> 32));
        const float d1 = __uint_as_float((u32)(G1 >> 32));
        const float d2 = __uint_as_float((u32)(G2 >> 32));
        // w = 1/(dist^2 + eps), normalized
        const float w0 = 1.0f / (d0 + EPS_W);
        const float w1 = 1.0f / (d1 + EPS_W);
        const float w2 = 1.0f / (d2 + EPS_W);
        const float inv = 1.0f / (w0 + w1 + w2);
        s_w[tid * 3 + 0] = w0 * inv;
        s_w[tid * 3 + 1] = w1 * inv;
        s_w[tid * 3 + 2] = w2 * inv;
        s_i[tid * 3 + 0] = (int)(u32)G0;
        s_i[tid * 3 + 1] = (int)(u32)G1;
        s_i[tid * 3 + 2] = (int)(u32)G2;
    }
    __syncthreads();

    // ---- Phase 4: coalesced weighted feature blend (memory-bound part)
    const float4* f4 = (const float4*)feats;
    float4*       o4 = (float4*)outp;
    for (int r = wave; r < TILE_M; r += WAVES) {
        const int row = blockIdx.x * TILE_M + r;
        const float w0 = s_w[r * 3 + 0];
        const float w1 = s_w[r * 3 + 1];
        const float w2 = s_w[r * 3 + 2];
        const float4* p0 = f4 + (size_t)s_i[r * 3 + 0] * (DIM / 4);
        const float4* p1 = f4 + (size_t)s_i[r * 3 + 1] * (DIM / 4);
        const float4* p2 = f4 + (size_t)s_i[r * 3 + 2] * (DIM / 4);
        float4*       po = o4 + (size_t)row * (DIM / 4);
#pragma unroll
        for (int c = lane; c < DIM / 4; c += 32) {
            const float4 a = p0[c];
            const float4 b = p1[c];
            const float4 g = p2[c];
            float4 acc;
            acc.x = w0 * a.x + w1 * b.x + w2 * g.x;
            acc.y = w0 * a.y + w1 * b.y + w2 * g.y;
            acc.z = w0 * a.z + w1 * b.z + w2 * g.z;
            acc.w = w0 * a.w + w1 * b.w + w2 * g.w;
            po[c] = acc;
        }
    }
}

extern "C" void kernel_launch(void* const* d_in, const int* in_sizes, int n_in,
                              void* d_out, int out_size, void* d_ws, size_t ws_size,
                              hipStream_t stream) {
    (void)in_sizes; (void)n_in; (void)d_ws; (void)ws_size; (void)out_size;
    const float* xyz_sub   = (const float*)d_in[0];  // [2,8,2048,3]
    const float* feats_sub = (const float*)d_in[1];  // [2,8,2048,256]
    const float* xyz_org   = (const float*)d_in[2];  // [2,8,8192,3]
    float*       out       = (float*)d_out;          // [2,8,8192,256]

    dim3 grid(NPTS / TILE_M, BLPAIRS);  // 64 x 16
    fp_knn_wmma_kernel<<<grid, dim3(BLOCK), 0, stream>>>(xyz_sub, feats_sub, xyz_org, out);
}